// FieldAwareFactorizationMachineModel_33208687133326
// MI455X (gfx1250) — compile-verified
//
#include <hip/hip_runtime.h>
#include <math.h>

// CDNA5 / gfx1250: wave32, WMMA 16x16x4 f32 (exact fp32 matrix op),
// async global->LDS gather (ASYNCcnt-tracked, no VGPR staging).
typedef float v2f __attribute__((ext_vector_type(2)));
typedef float v8f __attribute__((ext_vector_type(8)));

#define NF     23            // num fields
#define ED     16            // embed dim
#define NPAIRS 253           // NF*(NF-1)/2
#define NROWS  (NF * NF)     // 529 (row grid; diagonal rows never touched)
#define NGROWS (NF * (NF-1)) // 506 off-diagonal rows actually gathered
#define VTOT   205470        // total vocab
#define USE_ASYNC_LDS 1

__constant__ int c_offsets[NF] = {
    0, 1000, 51000, 71000, 81000, 86000, 88000, 89000, 89500,
    189500, 197500, 201500, 203500, 204500, 205000, 205250, 205350,
    205400, 205430, 205450, 205460, 205465, 205468};

// Compile-time tables: pair p -> (Lbase, Rbase) element offsets in s_rows,
// and the list of 506 off-diagonal rows to gather (packed row | i<<10 | j<<15).
struct Tables {
    unsigned int pairLR[256];
    unsigned int rows[NGROWS];
};
constexpr Tables make_tables() {
    Tables t{};
    int p = 0;
    for (int i = 0; i < NF; ++i)
        for (int j = i + 1; j < NF; ++j) {
            const unsigned L = (unsigned)((j * NF + i) * ED);  // E[i, idx_j, :]
            const unsigned R = (unsigned)((i * NF + j) * ED);  // E[j, idx_i, :]
            t.pairLR[p++] = L | (R << 16);
        }
    for (; p < 256; ++p)  // pad slots -> zero row (dot contributes 0)
        t.pairLR[p] = (unsigned)(NROWS * ED) | ((unsigned)(NROWS * ED) << 16);
    int n = 0;
    for (int j = 0; j < NF; ++j)
        for (int i = 0; i < NF; ++i)
            if (i != j)
                t.rows[n++] = (unsigned)(j * NF + i) | ((unsigned)i << 10) | ((unsigned)j << 15);
    return t;
}
__constant__ Tables c_tab = make_tables();

__global__ __launch_bounds__(128)
void ffm_kernel(const int* __restrict__ x, const float* __restrict__ E,
                const float* __restrict__ W, const float* __restrict__ bias,
                float* __restrict__ out)
{
    // row (j*NF+i) = E[i, idx_j, :]; row 529 = zeros (pad slots)
    __shared__ float s_rows[(NROWS + 1) * ED];   // 33,920 B
    __shared__ int   s_idx[NF];
    __shared__ float s_acc;

    const int b   = blockIdx.x;
    const int tid = threadIdx.x;

    if (tid == 0) s_acc = bias[0];
    if (tid < NF) s_idx[tid] = x[b * NF + tid] + c_offsets[tid];
    if (tid < ED) s_rows[NROWS * ED + tid] = 0.f;   // zero row
    __syncthreads();

    // linear term: sum_j W[idx_j]
    if (tid < NF) atomicAdd(&s_acc, W[s_idx[tid]]);

    // ---- Gather 506 off-diagonal rows x 64B into LDS: 4 lanes/row, 16B/lane ----
#if USE_ASYNC_LDS
    for (int t = tid; t < NGROWS * 4; t += 128) {
        const unsigned e = c_tab.rows[t >> 2];
        const int c   = t & 3;
        const int row = e & 1023;
        const int i   = (e >> 10) & 31;
        const int j   = (e >> 15) & 31;
        const float* src = E + ((size_t)i * VTOT + (size_t)s_idx[j]) * ED + c * 4;
        const unsigned long long gaddr = (unsigned long long)(const void*)src;
        // generic LDS pointer: addr[31:0] = workgroup-relative LDS byte offset
        const unsigned int ldsoff =
            (unsigned int)(size_t)(const void*)(s_rows + row * ED + c * 4);
        asm volatile("global_load_async_to_lds_b128 %0, %1, off"
                     :: "v"(ldsoff), "v"(gaddr) : "memory");
    }
    asm volatile("s_wait_asynccnt 0x0" ::: "memory");
#else
    for (int t = tid; t < NGROWS * 4; t += 128) {
        const unsigned e = c_tab.rows[t >> 2];
        const int c   = t & 3;
        const int row = e & 1023;
        const int i   = (e >> 10) & 31;
        const int j   = (e >> 15) & 31;
        const float4 v = *(const float4*)(E + ((size_t)i * VTOT + (size_t)s_idx[j]) * ED + c * 4);
        *(float4*)(s_rows + row * ED + c * 4) = v;
    }
#endif
    __syncthreads();

    // ---- Pair dot products via V_WMMA_F32_16X16X4_F32 ----
    // Pair slot m feeds A row m (L vector) and B col m (R vector); 4 chained
    // K=4 WMMAs cover K=0..15; diagonal of the shared C accumulator collects
    // one dot product per slot per group. Pad slots read the zero row.
    const int wave = tid >> 5;
    const int lane = tid & 31;
    const int m    = lane & 15;
    const int koff = (lane >> 4) << 1;   // lanes 0-15: K=k0..k0+1, lanes 16-31: +2

    v8f acc = {};
    #pragma unroll
    for (int q = 0; q < 4; ++q) {
        const unsigned e = c_tab.pairLR[(wave * 4 + q) * 16 + m];
        const float* Lp = s_rows + (e & 0xFFFFu) + koff;
        const float* Rp = s_rows + (e >> 16) + koff;
        #pragma unroll
        for (int k0 = 0; k0 < ED; k0 += 4) {
            const v2f a  = *(const v2f*)(Lp + k0);
            const v2f bb = *(const v2f*)(Rp + k0);
            acc = __builtin_amdgcn_wmma_f32_16x16x4_f32(
                false, a, false, bb, (short)0, acc, false, false);
        }
    }

    // Diagonal of C: (m,m) at lane m / vgpr m (m<8), lane m+16 / vgpr m-8 (m>=8)
    float d = 0.f;
    #pragma unroll
    for (int k = 0; k < 8; ++k)
        d += ((lane == k) || (lane == 24 + k)) ? acc[k] : 0.f;

    // wave32 reduction
    #pragma unroll
    for (int off = 16; off > 0; off >>= 1)
        d += __shfl_xor(d, off, 32);

    if (lane == 0) atomicAdd(&s_acc, d);
    __syncthreads();

    if (tid == 0) {
        const float z = s_acc;
        out[b] = 1.0f / (1.0f + expf(-z));
    }
}

extern "C" void kernel_launch(void* const* d_in, const int* in_sizes, int n_in,
                              void* d_out, int out_size, void* d_ws, size_t ws_size,
                              hipStream_t stream) {
    const int*   x    = (const int*)  d_in[0];   // [B, 23] indices
    const float* E    = (const float*)d_in[1];   // [23, 205470, 16]
    const float* W    = (const float*)d_in[2];   // [205470]
    const float* bias = (const float*)d_in[3];   // [1]
    float* out = (float*)d_out;                  // [B]
    ffm_kernel<<<out_size, 128, 0, stream>>>(x, E, W, bias, out);
}